// Bayes_4294967296643
// MI455X (gfx1250) — compile-verified
//
#include <hip/hip_runtime.h>
#include <math.h>

// ---------------------------------------------------------------------------
// Bayes attention module for MI455X (gfx1250, wave32, WMMA + TDM).
// Heavy op = 3x3 conv (512->256 ch) = 38.7 GFLOP fp32 -> implicit GEMM via
// V_WMMA_F32_16X16X4_F32, inputs staged by Tensor Data Mover (double-buffered
// LDS, TENSORcnt pipelining). Attention / softmax / reductions are VALU.
// ---------------------------------------------------------------------------

typedef __attribute__((ext_vector_type(2))) float v2f;
typedef __attribute__((ext_vector_type(8))) float v8f;
typedef __attribute__((ext_vector_type(4))) unsigned int u32x4;
typedef __attribute__((ext_vector_type(4))) int i32x4;
typedef __attribute__((ext_vector_type(8))) int i32x8;

#define NB     4
#define CCH    256
#define HH     64
#define WW     64
#define HWPIX  (HH * WW)          // 4096
#define LTOT   (NB * HWPIX)       // 16384
#define NCLASS 20
#define HEADS  4
#define HDIM   (CCH / HEADS)      // 64

#define PW     66                 // padded width/height of conv input
#define PIMG   (PW * PW)          // 4356
#define C2     (2 * CCH)          // 512 conv input channels
#define CI_CHUNK 16
#define NCHUNK (C2 / CI_CHUNK)    // 32

#if defined(__has_builtin)
#if __has_builtin(__builtin_amdgcn_tensor_load_to_lds) && \
    __has_builtin(__builtin_amdgcn_s_wait_tensorcnt)
#define USE_TDM 1
#endif
#endif
#ifndef USE_TDM
#define USE_TDM 0
#endif

// ------------------------- channel mean (per n,c) --------------------------
__global__ void chan_mean_kernel(const float* __restrict__ src, float* __restrict__ out) {
    __shared__ float red[256];
    int nc = blockIdx.x;
    const float* p = src + (size_t)nc * HWPIX;
    float s = 0.f;
    for (int i = threadIdx.x; i < HWPIX; i += 256) s += p[i];
    red[threadIdx.x] = s;
    __syncthreads();
    for (int off = 128; off > 0; off >>= 1) {
        if ((int)threadIdx.x < off) red[threadIdx.x] += red[threadIdx.x + off];
        __syncthreads();
    }
    if (threadIdx.x == 0) out[nc] = red[0] * (1.0f / HWPIX);
}

__global__ void init_max_kernel(unsigned int* mb) {
    if (threadIdx.x < NB) mb[threadIdx.x] = 0u;
}

// --------- label-class heatmap (ReLU) + per-sample max (float bits) --------
__global__ void mask_kernel(const float* __restrict__ feats, const float* __restrict__ w_gap,
                            const int* __restrict__ label, float* __restrict__ mask_raw,
                            unsigned int* __restrict__ maxbits) {
    __shared__ float wv[CCH];
    int n = blockIdx.x >> 4;
    int p = ((blockIdx.x & 15) << 8) + threadIdx.x;
    int lab = label[n];
    for (int c = threadIdx.x; c < CCH; c += blockDim.x) wv[c] = w_gap[lab * CCH + c];
    __syncthreads();
    const float* f = feats + (size_t)n * CCH * HWPIX + p;
    float acc = 0.f;
    for (int c = 0; c < CCH; ++c) acc += f[(size_t)c * HWPIX] * wv[c];
    float m = fmaxf(acc, 0.f);
    mask_raw[n * HWPIX + p] = m;
    atomicMax(&maxbits[n], __float_as_uint(m));
}

__global__ void mask_norm_kernel(float* __restrict__ mask, const unsigned int* __restrict__ maxbits) {
    int n = blockIdx.x >> 4;
    int p = ((blockIdx.x & 15) << 8) + threadIdx.x;
    float mx = fmaxf(__uint_as_float(maxbits[n]), 1.0f);
    mask[n * HWPIX + p] /= mx;
}

// --------------------- pred[n,k] = w[k,:] . chan_mean[n,:] -----------------
__global__ void pred_kernel(const float* __restrict__ cm, const float* __restrict__ wmat,
                            float* __restrict__ out) {
    int i = threadIdx.x;
    if (i >= NB * NCLASS) return;
    int n = i / NCLASS, k = i % NCLASS;
    float acc = 0.f;
    for (int c = 0; c < CCH; ++c) acc += wmat[k * CCH + c] * cm[n * CCH + c];
    out[i] = acc;
}

// ----------------- windowed attention scores + softmax ---------------------
template <int KS, int DIL>
__global__ void att_kernel(const float* __restrict__ Kc, const float* __restrict__ Qc,
                           float* __restrict__ att) {
    constexpr int K2 = KS * KS;
    constexpr int R  = KS / 2;
    int idx = blockIdx.x * blockDim.x + threadIdx.x;
    int l = idx & (LTOT - 1);
    int a = idx >> 14;
    int n = l >> 12, y = (l >> 6) & 63, x = l & 63;

    int off[K2];
    bool ok[K2];
#pragma unroll
    for (int k = 0; k < K2; ++k) {
        int fi = k / KS, fj = k % KS;
        int yy = y + (fi - R) * DIL;
        int xx = x + (fj - R) * DIL;
        ok[k]  = (yy >= 0 && yy < HH && xx >= 0 && xx < WW);
        off[k] = yy * WW + xx;
    }
    float s[K2];
#pragma unroll
    for (int k = 0; k < K2; ++k) s[k] = 0.f;

    const float* qb = Qc + ((size_t)(n * CCH + a * HDIM)) * HWPIX + y * WW + x;
    const float* kb = Kc + ((size_t)(n * CCH + a * HDIM)) * HWPIX;
    for (int hd = 0; hd < HDIM; ++hd) {
        float qv = qb[(size_t)hd * HWPIX];
        const float* kc = kb + (size_t)hd * HWPIX;
#pragma unroll
        for (int k = 0; k < K2; ++k)
            if (ok[k]) s[k] += kc[off[k]] * qv;   // zero-padded taps keep score 0
    }
    float mx = s[0];
#pragma unroll
    for (int k = 1; k < K2; ++k) mx = fmaxf(mx, s[k]);
    float den = 0.f;
#pragma unroll
    for (int k = 0; k < K2; ++k) { s[k] = __expf(s[k] - mx); den += s[k]; }
    float inv = 1.0f / den;
#pragma unroll
    for (int k = 0; k < K2; ++k) att[((size_t)(a * K2 + k)) * LTOT + l] = s[k] * inv;
}

// --------- infusion output: weighted 5x5(dil3) gather of feats -------------
__global__ void infusion_out_kernel(const float* __restrict__ feats, const float* __restrict__ att,
                                    float* __restrict__ feats_r) {
    int idx = blockIdx.x * blockDim.x + threadIdx.x;
    int pix = idx & (HWPIX - 1);
    int c   = (idx >> 12) & (CCH - 1);
    int n   = idx >> 20;
    int y = pix >> 6, x = pix & 63;
    int a = c >> 6;
    int l = n * HWPIX + pix;
    const float* f = feats + ((size_t)(n * CCH + c)) * HWPIX;
    float acc = 0.f;
#pragma unroll
    for (int k = 0; k < 25; ++k) {
        int fi = k / 5, fj = k % 5;
        int yy = y + (fi - 2) * 3, xx = x + (fj - 2) * 3;
        if (yy >= 0 && yy < HH && xx >= 0 && xx < WW)
            acc += att[((size_t)(a * 25 + k)) * LTOT + l] * f[yy * WW + xx];
    }
    feats_r[idx] = acc;
}

// ------------------------- xpad construction -------------------------------
// xpad: (NB, 512, 66, 66) zero-bordered conv input = concat(feats*mask, feats_d)
__global__ void zero4_kernel(float* __restrict__ p) {
    int i = blockIdx.x * blockDim.x + threadIdx.x;
    ((float4*)p)[i] = make_float4(0.f, 0.f, 0.f, 0.f);
}

__global__ void fill_xpad_feats_kernel(const float* __restrict__ feats,
                                       const float* __restrict__ mask,
                                       float* __restrict__ xpad) {
    int idx = blockIdx.x * blockDim.x + threadIdx.x;   // NB*CCH*HWPIX
    int pix = idx & (HWPIX - 1);
    int c   = (idx >> 12) & (CCH - 1);
    int n   = idx >> 20;
    int y = pix >> 6, x = pix & 63;
    float v = feats[idx] * mask[n * HWPIX + pix];
    xpad[((size_t)(n * C2 + c) * PW + (y + 1)) * PW + (x + 1)] = v;
}

// ---- diffusion as gather (col2im inverted) writing xpad channels 256..511 -
__global__ void diffusion_kernel(const float* __restrict__ feats, const float* __restrict__ att,
                                 const float* __restrict__ mask, float* __restrict__ xpad) {
    int idx = blockIdx.x * blockDim.x + threadIdx.x;
    int pix = idx & (HWPIX - 1);
    int c   = (idx >> 12) & (CCH - 1);
    int n   = idx >> 20;
    int y = pix >> 6, x = pix & 63;
    int a = c >> 6;
    const float* f = feats + ((size_t)(n * CCH + c)) * HWPIX;
    float acc = 0.f;
#pragma unroll
    for (int fi = 0; fi < 3; ++fi) {
#pragma unroll
        for (int fj = 0; fj < 3; ++fj) {
            int h = y + 6 - 6 * fi, w = x + 6 - 6 * fj;
            if (h >= 0 && h < HH && w >= 0 && w < WW) {
                int lsrc = n * HWPIX + h * WW + w;
                int k = fi * 3 + fj;
                acc += att[((size_t)(a * 9 + k)) * LTOT + lsrc] * mask[lsrc] * f[h * WW + w];
            }
        }
    }
    xpad[((size_t)(n * C2 + CCH + c) * PW + (y + 1)) * PW + (x + 1)] = acc;
}

// ------------------------- TDM helpers -------------------------------------
#if USE_TDM
__device__ __forceinline__ unsigned lds_off_of(const void* p) {
    return (unsigned)(unsigned long long)(const __attribute__((address_space(3))) char*)p;
}

// 2D tile load: tile_d1 lines x tile_d0 contiguous fp32 elems, line stride
// stride0 elems.  D# packed per CDNA5 ISA 8.3/8.4 (count=1, type=2, 4B data).
__device__ __forceinline__ void tdm_load_2d(unsigned lds_byte_off, const float* gptr,
                                            unsigned tile_d0, unsigned tile_d1,
                                            unsigned stride0) {
    unsigned long long ga = (unsigned long long)gptr;
    u32x4 g0;
    g0.x = 1u;                                            // count=1, user, no gather
    g0.y = lds_byte_off;                                  // lds_addr
    g0.z = (unsigned)(ga & 0xFFFFFFFFull);                // global_addr[31:0]
    g0.w = (unsigned)((ga >> 32) & 0x1FFFFFFull) | (2u << 30);  // ga[56:32] | type=2
    i32x8 g1;
    g1.s0 = (int)(2u << 16);                              // wg_mask=0, data_size=4B
    g1.s1 = (int)((tile_d0 & 0xFFFFu) << 16);             // tensor_dim0[15:0]
    g1.s2 = (int)(((tile_d0 >> 16) & 0xFFFFu) | ((tile_d1 & 0xFFFFu) << 16)); // td0 hi | td1 lo
    g1.s3 = (int)(((tile_d1 >> 16) & 0xFFFFu) | ((tile_d0 & 0xFFFFu) << 16)); // td1 hi | tile_dim0
    g1.s4 = (int)(tile_d1 & 0xFFFFu);                     // tile_dim1 | tile_dim2=0
    g1.s5 = (int)stride0;                                 // tensor_dim0_stride[31:0]
    g1.s6 = 0;
    g1.s7 = 0;
    i32x4 z4 = {0, 0, 0, 0};
#if __clang_major__ >= 23
    i32x8 z8 = {0, 0, 0, 0, 0, 0, 0, 0};
    __builtin_amdgcn_tensor_load_to_lds(g0, g1, z4, z4, z8, 0);
#else
    __builtin_amdgcn_tensor_load_to_lds(g0, g1, z4, z4, 0);
#endif
}
#endif

// ------------------- conv3x3 (512->256) via fp32 WMMA ----------------------
// Implicit GEMM: block = 4 waves; each wave: 16 couts x 64 pixels (4 D tiles),
// K = 512 cin x 9 taps in CI_CHUNK slices, double-buffered LDS fed by TDM.
__global__ __launch_bounds__(128) void conv_wmma_kernel(
    const float* __restrict__ xpad, const float* __restrict__ Wc,
    const float* __restrict__ bias, float* __restrict__ out) {
    __shared__ float ldsX[2][CI_CHUNK * 198];        // 16 ch x (3 rows x 66 cols)
    __shared__ float ldsW[2][64 * CI_CHUNK * 9];     // 64 couts x 16 ci x 9 taps

    int tid  = threadIdx.x;
    int lane = tid & 31;
    int wave = tid >> 5;
    int p0 = blockIdx.x << 6;                        // 64 pixels: one full (n,y) row
    int n  = p0 >> 12, y = (p0 >> 6) & 63;
    int co0b = blockIdx.y << 6;

    int half = lane >> 4;                            // K-half per ISA A/B fp32 layout
    int mrow = lane & 15;                            // A: M index; B: pixel index
    int kb   = half << 1;

    v8f acc0 = {0.f,0.f,0.f,0.f,0.f,0.f,0.f,0.f};
    v8f acc1 = acc0, acc2 = acc0, acc3 = acc0;

    const float* xbase = xpad + ((size_t)n * C2 * PW + y) * PW;   // + ci*PIMG
    const float* wbase = Wc + (size_t)co0b * (C2 * 9);            // + co*4608 + cc*9

#if USE_TDM
    if (wave == 0)
        tdm_load_2d(lds_off_of(&ldsX[0][0]), xbase, 198, CI_CHUNK, PIMG),
        tdm_load_2d(lds_off_of(&ldsW[0][0]), wbase, CI_CHUNK * 9, 64, C2 * 9);
#endif

    for (int cc = 0; cc < NCHUNK; ++cc) {
        int pb = cc & 1;
#if USE_TDM
        if (wave == 0) {
            if (cc + 1 < NCHUNK) {
                tdm_load_2d(lds_off_of(&ldsX[(cc + 1) & 1][0]),
                            xbase + (size_t)(cc + 1) * CI_CHUNK * PIMG,
                            198, CI_CHUNK, PIMG);
                tdm_load_2d(lds_off_of(&ldsW[(cc + 1) & 1][0]),
                            wbase + (size_t)(cc + 1) * CI_CHUNK * 9,
                            CI_CHUNK * 9, 64, C2 * 9);
                __builtin_amdgcn_s_wait_tensorcnt(2);   // chunk cc's pair done
            } else {
                __builtin_amdgcn_s_wait_tensorcnt(0);
            }
        }
#else
        for (int i = tid; i < CI_CHUNK * 198; i += 128)
            ldsX[pb][i] = xbase[(size_t)(cc * CI_CHUNK + i / 198) * PIMG + (i % 198)];
        for (int i = tid; i < 64 * CI_CHUNK * 9; i += 128)
            ldsW[pb][i] = wbase[(size_t)(i / (CI_CHUNK * 9)) * (C2 * 9) + cc * CI_CHUNK * 9 +
                                (i % (CI_CHUNK * 9))];
#endif
        __syncthreads();

        const float* lw = &ldsW[pb][(wave << 4 | mrow) * (CI_CHUNK * 9)];
        const float* lx = &ldsX[pb][mrow];
#pragma unroll
        for (int kpos = 0; kpos < 9; ++kpos) {
            int fi = kpos / 3, fj = kpos % 3;
            const float* lxk = lx + fi * 66 + fj;     // + (ci)*198 + t*16
#pragma unroll
            for (int ci = 0; ci < CI_CHUNK; ci += 4) {
                v2f a, b;
                a.x = lw[(ci + kb) * 9 + kpos];       // A 16x4: lane=M+16*(K>=2)
                a.y = lw[(ci + kb + 1) * 9 + kpos];
                const float* bp = lxk + (ci + kb) * 198;
                b.x = bp[0];  b.y = bp[198];
                acc0 = __builtin_amdgcn_wmma_f32_16x16x4_f32(false, a, false, b, (short)0, acc0, false, false);
                b.x = bp[16]; b.y = bp[198 + 16];
                acc1 = __builtin_amdgcn_wmma_f32_16x16x4_f32(false, a, false, b, (short)0, acc1, false, false);
                b.x = bp[32]; b.y = bp[198 + 32];
                acc2 = __builtin_amdgcn_wmma_f32_16x16x4_f32(false, a, false, b, (short)0, acc2, false, false);
                b.x = bp[48]; b.y = bp[198 + 48];
                acc3 = __builtin_amdgcn_wmma_f32_16x16x4_f32(false, a, false, b, (short)0, acc3, false, false);
            }
        }
        __syncthreads();
    }

    // epilogue: documented 16x16 fp32 C/D layout + bias
    v8f accs[4] = {acc0, acc1, acc2, acc3};
#pragma unroll
    for (int t = 0; t < 4; ++t) {
        int x = (t << 4) + mrow;
#pragma unroll
        for (int r = 0; r < 8; ++r) {
            int co = co0b + (wave << 4) + r + (half << 3);
            out[((size_t)(n * CCH + co)) * HWPIX + y * WW + x] = accs[t][r] + bias[co];
        }
    }
}

// ---------------------------------------------------------------------------
extern "C" void kernel_launch(void* const* d_in, const int* in_sizes, int n_in,
                              void* d_out, int out_size, void* d_ws, size_t ws_size,
                              hipStream_t stream) {
    const float* feats   = (const float*)d_in[0];
    const float* Kin     = (const float*)d_in[1];
    const float* Qin     = (const float*)d_in[2];
    const float* w_gap   = (const float*)d_in[3];
    const float* w_gap_r = (const float*)d_in[4];
    const float* conv_w  = (const float*)d_in[5];
    const float* conv_b  = (const float*)d_in[6];
    const int*   label   = (const int*)d_in[7];

    float* out_nxt    = (float*)d_out;
    float* out_pred   = out_nxt + (size_t)NB * CCH * HWPIX;
    float* out_pred_r = out_pred + NB * NCLASS;

    float* ws        = (float*)d_ws;
    float* cm        = ws;                                    // 1024
    float* cm_r      = ws + 1024;                             // 1024
    unsigned int* mb = (unsigned int*)(ws + 2048);            // 4 (+pad)
    float* maskb     = ws + 2048 + 64;                        // 16384
    float* att_inf   = maskb + NB * HWPIX;                    // 4*25*16384
    float* att_dif   = att_inf + (size_t)HEADS * 25 * LTOT;   // 4*9*16384
    float* feats_r   = att_dif + (size_t)HEADS * 9 * LTOT;    // 4M
    float* xpad      = feats_r + (size_t)NB * CCH * HWPIX;    // 4*512*66*66

    // pred + mask
    chan_mean_kernel<<<NB * CCH, 256, 0, stream>>>(feats, cm);
    init_max_kernel<<<1, 32, 0, stream>>>(mb);
    mask_kernel<<<NB * (HWPIX / 256), 256, 0, stream>>>(feats, w_gap, label, maskb, mb);
    mask_norm_kernel<<<NB * (HWPIX / 256), 256, 0, stream>>>(maskb, mb);
    pred_kernel<<<1, 128, 0, stream>>>(cm, w_gap, out_pred);

    // infusion (reference swap: Kc=Q, Qc=K) -> feats_r -> pred_r
    att_kernel<5, 3><<<(HEADS * LTOT) / 256, 256, 0, stream>>>(Qin, Kin, att_inf);
    infusion_out_kernel<<<(NB * CCH * HWPIX) / 256, 256, 0, stream>>>(feats, att_inf, feats_r);
    chan_mean_kernel<<<NB * CCH, 256, 0, stream>>>(feats_r, cm_r);
    pred_kernel<<<1, 128, 0, stream>>>(cm_r, w_gap_r, out_pred_r);

    // build zero-padded conv input: channels 0..255 = feats*mask, 256..511 = diffusion
    zero4_kernel<<<(NB * C2 * PIMG) / 4 / 256, 256, 0, stream>>>(xpad);
    fill_xpad_feats_kernel<<<(NB * CCH * HWPIX) / 256, 256, 0, stream>>>(feats, maskb, xpad);
    att_kernel<3, 6><<<(HEADS * LTOT) / 256, 256, 0, stream>>>(Kin, Qin, att_dif);
    diffusion_kernel<<<(NB * CCH * HWPIX) / 256, 256, 0, stream>>>(feats, att_dif, maskb, xpad);

    // conv3x3 via fp32 WMMA implicit GEMM, TDM-fed double-buffered LDS
    dim3 cgrid(LTOT / 64, CCH / 64);
    conv_wmma_kernel<<<cgrid, 128, 0, stream>>>(xpad, conv_w, conv_b, out_nxt);
}